// Attention_54846732370143
// MI455X (gfx1250) — compile-verified
//
#include <hip/hip_runtime.h>
#include <math.h>

// ---------------------------------------------------------------------------
// MI455X (gfx1250, wave32) decode-attention pipeline.
// HBM floor: ~1.2 GB traffic (KV cache dominates) @ 23.3 TB/s ~= 51 us.
// - Dense GEMMs: V_WMMA_F32_16X16X4_F32 (qkv proj needs ~390 TFLOP/s fp32 at
//   bandwidth limit -> VALU insufficient, f32 WMMA required).
// - KV stream: GLOBAL_LOAD_ASYNC_TO_LDS_B128 double-buffered per wave,
//   ASYNCcnt-tracked, so the 1 GB stream overlaps compute without VGPR cost.
// ---------------------------------------------------------------------------

typedef float v2f __attribute__((ext_vector_type(2)));
typedef float v8f __attribute__((ext_vector_type(8)));
typedef int   v4i_vs __attribute__((vector_size(4 * sizeof(int))));

#define B        32
#define DIM      4096
#define CACHE_LEN 4096
#define HD       128
#define NH       32
#define NKV      8
#define HPG      4               // heads per KV group (GQA 4:1)
#define QKV_OUT  6144            // (32 + 2*8) * 128
#define SPLITK   8
#define NCHUNK   8
#define CHUNK    (CACHE_LEN / NCHUNK)   // 512 keys per block
#define KEYS_PER_WAVE (CHUNK / 8)       // 64 contiguous keys per wave
#define TILE_KEYS 4                      // keys per async tile
#define PART_STRIDE 520                  // 2 stats + 512 values + pad

// ---- gfx1250 async global->LDS copy (guarded; falls back to sync loads) ----
#if defined(__HIP_DEVICE_COMPILE__) && \
    __has_builtin(__builtin_amdgcn_global_load_async_to_lds_b128)
#define HAVE_ASYNC_LDS 1
#else
#define HAVE_ASYNC_LDS 0
#endif

__device__ __forceinline__ void async_copy16(const float* g, float* l) {
#if HAVE_ASYNC_LDS
  // ROCm 7.2 signature: (v4i __device__* src, v4i __shared__* dst, off, cpol)
  __builtin_amdgcn_global_load_async_to_lds_b128(
      (__attribute__((address_space(1))) v4i_vs*)g,
      (__attribute__((address_space(3))) v4i_vs*)l, 0, 0);
#else
  *(float4*)l = *(const float4*)g;
#endif
}

__device__ __forceinline__ void async_wait_prev_tile() {   // <= 8 outstanding
#if HAVE_ASYNC_LDS
#if __has_builtin(__builtin_amdgcn_s_wait_asynccnt)
  __builtin_amdgcn_s_wait_asynccnt(8);
#else
  asm volatile("s_wait_asynccnt 8" ::: "memory");
#endif
#endif
  asm volatile("" ::: "memory");
}

__device__ __forceinline__ void async_wait_all() {
#if HAVE_ASYNC_LDS
#if __has_builtin(__builtin_amdgcn_s_wait_asynccnt)
  __builtin_amdgcn_s_wait_asynccnt(0);
#else
  asm volatile("s_wait_asynccnt 0" ::: "memory");
#endif
#endif
  asm volatile("" ::: "memory");
}

__global__ void zero_f32(float* __restrict__ p, int n) {
  int i = blockIdx.x * blockDim.x + threadIdx.x;
  if (i < n) p[i] = 0.0f;
}

// ---------------------------------------------------------------------------
// C[m,n] += sum_k X[m,k] * W[n,k], M == 32 (batch), via WMMA f32 16x16x4.
// Block = 128 threads = 4 waves; wave owns a 32(M) x 16(N) C tile.
// Uniform-base + 32-bit lane offsets -> saddr-form global loads.
// Fragment layouts (ISA 7.12.2, wave32):
//   A 16x4 : lane<16 -> {A[m][k+0],A[m][k+1]}, lane>=16 -> {A[m][k+2],A[m][k+3]}
//   B 4x16 : lane<16 -> {B[k+0][n],B[k+1][n]}, lane>=16 -> {B[k+2][n],B[k+3][n]}
//   C 16x16: vgpr r, lane<16 -> (M=r, N=lane), lane>=16 -> (M=8+r, N=lane-16)
// ---------------------------------------------------------------------------
__global__ void gemm_xt_wmma(const float* __restrict__ X,
                             const float* __restrict__ W,
                             float* __restrict__ C,
                             int N, int K) {
  const int wave = threadIdx.x >> 5;
  const int lane = threadIdx.x & 31;
  const int half = lane >> 4;           // 0: K pair {0,1}; 1: K pair {2,3}
  const int lm   = lane & 15;
  const int n    = blockIdx.x * 64 + wave * 16 + lm;
  const int kc   = blockIdx.y * (K / SPLITK);
  const int kend = kc + (K / SPLITK);

  // 32-bit element offsets from uniform bases (max 25M elements < 2^31 bytes).
  const uint32_t a0o = (uint32_t)lm * (uint32_t)K;
  const uint32_t a1o = (uint32_t)(16 + lm) * (uint32_t)K;
  const uint32_t wo  = (uint32_t)n * (uint32_t)K;

  v8f c0 = {};
  v8f c1 = {};

  for (int k = kc; k < kend; k += 32) {
    // Stream the weight row ahead (global_prefetch_b8).
    __builtin_prefetch(W + wo + (uint32_t)(k + 2 * half + 512), 0, 1);
#pragma unroll
    for (int kk = 0; kk < 32; kk += 4) {
      const uint32_t kb = (uint32_t)(k + kk + 2 * half);   // 8B aligned
      v2f a0 = *(const v2f*)(X + a0o + kb);
      v2f a1 = *(const v2f*)(X + a1o + kb);
      v2f bw = *(const v2f*)(W + wo + kb);
      c0 = __builtin_amdgcn_wmma_f32_16x16x4_f32(false, a0, false, bw,
                                                 (short)0, c0, false, false);
      c1 = __builtin_amdgcn_wmma_f32_16x16x4_f32(false, a1, false, bw,
                                                 (short)0, c1, false, false);
    }
  }

#pragma unroll
  for (int r = 0; r < 8; ++r) {
    const int m = r + 8 * half;
    atomicAdd(&C[(size_t)m * N + n], c0[r]);
    atomicAdd(&C[(size_t)(16 + m) * N + n], c1[r]);
  }
}

// ---------------------------------------------------------------------------
// RoPE q (in place in xqkv) and k (into knew); copy v into vnew.
// Input caches are never mutated; the t == ctx_len row lives in scratch.
// ---------------------------------------------------------------------------
__global__ void rope_cache_kernel(float* __restrict__ xqkv,
                                  float* __restrict__ knew,
                                  float* __restrict__ vnew,
                                  const int* __restrict__ ctx_len_p) {
  const int i = blockIdx.x * blockDim.x + threadIdx.x;
  const float pos = (float)(*ctx_len_p);
  const float LOG_THETA = 9.210340371976184f;   // ln(10000)

  const int QP = B * NH * (HD / 2);
  const int KP = B * NKV * (HD / 2);
  const int VE = B * NKV * HD;

  if (i < QP) {
    const int b  = i / (NH * HD / 2);
    const int r  = i % (NH * HD / 2);
    const int h  = r / (HD / 2);
    const int pi = r % (HD / 2);
    const float f = __expf(-(2.0f * pi / HD) * LOG_THETA);
    float sn, cs;
    __sincosf(pos * f, &sn, &cs);
    float* p = xqkv + (size_t)b * QKV_OUT + h * HD + 2 * pi;
    const float x1 = p[0], x2 = p[1];
    p[0] = x1 * cs - x2 * sn;
    p[1] = x1 * sn + x2 * cs;
  } else if (i < QP + KP) {
    const int j  = i - QP;
    const int b  = j / (NKV * HD / 2);
    const int r  = j % (NKV * HD / 2);
    const int g  = r / (HD / 2);
    const int pi = r % (HD / 2);
    const float f = __expf(-(2.0f * pi / HD) * LOG_THETA);
    float sn, cs;
    __sincosf(pos * f, &sn, &cs);
    const float* p = xqkv + (size_t)b * QKV_OUT + NH * HD + g * HD + 2 * pi;
    const float x1 = p[0], x2 = p[1];
    float* q = knew + ((size_t)b * NKV + g) * HD + 2 * pi;
    q[0] = x1 * cs - x2 * sn;
    q[1] = x1 * sn + x2 * cs;
  } else if (i < QP + KP + VE) {
    const int j = i - (QP + KP);
    const int b = j / (NKV * HD);
    const int r = j % (NKV * HD);
    vnew[(size_t)b * NKV * HD + r] =
        xqkv[(size_t)b * QKV_OUT + NH * HD + NKV * HD + r];
  }
}

// ---------------------------------------------------------------------------
// Flash-decode partial: grid (chunk, kv_head, batch), 256 threads = 8 waves.
// Wave w owns 64 contiguous keys; K/V rows stream through LDS via async
// global->LDS b128 copies (8 per 4-key tile), double-buffered against compute.
// Lane owns dims 4*lane..4*lane+3; 4-head GQA group online softmax with
// wave32 __shfl_xor reductions; 8 waves merged via LDS at the end.
// ---------------------------------------------------------------------------
__global__ void attn_partial_kernel(const float* __restrict__ xqkv,   // roped q
                                    const float* __restrict__ cache_k,
                                    const float* __restrict__ cache_v,
                                    const float* __restrict__ knew,
                                    const float* __restrict__ vnew,
                                    float* __restrict__ part,
                                    const int* __restrict__ ctx_len_p) {
  const int chunk = blockIdx.x;
  const int g     = blockIdx.y;
  const int b     = blockIdx.z;
  const int tid   = threadIdx.x;
  const int wave  = tid >> 5;
  const int lane  = tid & 31;
  const int ctx   = *ctx_len_p;
  const int T     = ctx + 1;
  const float scale = 0.08838834764831845f;     // 1/sqrt(128)

  // Per-wave double-buffered K/V tiles: [buf][key][K|V][dim] -> 64 KB total.
  __shared__ float tiles[8][2][TILE_KEYS][2][HD];

  const float* krow_new = knew + ((size_t)b * NKV + g) * HD;
  const float* vrow_new = vnew + ((size_t)b * NKV + g) * HD;
  const size_t cbase = ((size_t)b * CACHE_LEN * NKV + g) * HD;

  // q registers: q[h][j] = q_{g*HPG+h}[4*lane + j]
  float q[HPG][4];
#pragma unroll
  for (int h = 0; h < HPG; ++h) {
    const float4 qv = *(const float4*)(xqkv + (size_t)b * QKV_OUT +
                                       (g * HPG + h) * HD + 4 * lane);
    q[h][0] = qv.x; q[h][1] = qv.y; q[h][2] = qv.z; q[h][3] = qv.w;
  }

  const int base_w = chunk * CHUNK + wave * KEYS_PER_WAVE;
  const int NTILES = KEYS_PER_WAVE / TILE_KEYS;   // 16

  auto issue_tile = [&](int buf, int i) {
#pragma unroll
    for (int r = 0; r < TILE_KEYS; ++r) {
      const int t = base_w + TILE_KEYS * i + r;
      const float* ks;
      const float* vs;
      if (t < ctx) {
        const size_t off = cbase + (size_t)t * (NKV * HD);
        ks = cache_k + off;
        vs = cache_v + off;
      } else {                 // t == ctx -> staged new row; t >= T -> dummy
        ks = krow_new;
        vs = vrow_new;
      }
      async_copy16(ks + 4 * lane, &tiles[wave][buf][r][0][4 * lane]);
      async_copy16(vs + 4 * lane, &tiles[wave][buf][r][1][4 * lane]);
    }
  };

  float m = -INFINITY, l = 0.0f;
  float acc[HPG][4] = {};

  issue_tile(0, 0);
  for (int i = 0; i < NTILES; ++i) {
    if (i + 1 < NTILES) {
      issue_tile((i + 1) & 1, i + 1);
      async_wait_prev_tile();            // previous tile's 8 copies complete
    } else {
      async_wait_all();
    }

#pragma unroll
    for (int r = 0; r < TILE_KEYS; ++r) {
      const int t = base_w + TILE_KEYS * i + r;
      if (t >= T) break;                  // uniform per wave
      const float4 kv = *(const float4*)&tiles[wave][i & 1][r][0][4 * lane];
      const float4 vv = *(const float4*)&tiles[wave][i & 1][r][1][4 * lane];

      float s[HPG];
#pragma unroll
      for (int h = 0; h < HPG; ++h) {
        float p = q[h][0] * kv.x + q[h][1] * kv.y +
                  q[h][2] * kv.z + q[h][3] * kv.w;
#pragma unroll
        for (int o = 16; o > 0; o >>= 1) p += __shfl_xor(p, o, 32);
        s[h] = p * scale;
      }

      float mnew = m;
#pragma unroll
      for (int h = 0; h < HPG; ++h) mnew = fmaxf(mnew, s[h]);
      const float alpha = __expf(m - mnew);

      float psum = 0.0f;
#pragma unroll
      for (int h = 0; h < HPG; ++h) {
        const float p = __expf(s[h] - mnew);
        psum += p;
        acc[h][0] = acc[h][0] * alpha + p * vv.x;
        acc[h][1] = acc[h][1] * alpha + p * vv.y;
        acc[h][2] = acc[h][2] * alpha + p * vv.z;
        acc[h][3] = acc[h][3] * alpha + p * vv.w;
      }
      l = l * alpha + psum;
      m = mnew;
    }
  }

  // Merge 8 waves through LDS into one per-block partial record.
  __shared__ float sm[8];
  __shared__ float sl[8];
  __shared__ float sacc[8][HPG * HD];
  if (lane == 0) { sm[wave] = m; sl[wave] = l; }
#pragma unroll
  for (int h = 0; h < HPG; ++h) {
#pragma unroll
    for (int j = 0; j < 4; ++j) sacc[wave][h * HD + 4 * lane + j] = acc[h][j];
  }
  __syncthreads();

  float M = -INFINITY;
#pragma unroll
  for (int w = 0; w < 8; ++w) if (sl[w] > 0.0f) M = fmaxf(M, sm[w]);

  const size_t base = (size_t)((b * NKV + g) * NCHUNK + chunk) * PART_STRIDE;
  for (int e = tid; e < HPG * HD; e += 256) {
    float o = 0.0f;
#pragma unroll
    for (int w = 0; w < 8; ++w)
      if (sl[w] > 0.0f) o += sacc[w][e] * __expf(sm[w] - M);
    part[base + 2 + e] = o;
  }
  if (tid == 0) {
    float L = 0.0f;
#pragma unroll
    for (int w = 0; w < 8; ++w) if (sl[w] > 0.0f) L += sl[w] * __expf(sm[w] - M);
    part[base + 0] = M;
    part[base + 1] = L;
  }
}

// ---------------------------------------------------------------------------
// Combine NCHUNK partials per (b, kv_head), normalize, write attn output in
// (b, [g*HPG + h]*HD + d) layout so the final GEMM consumes it directly.
// ---------------------------------------------------------------------------
__global__ void attn_reduce_kernel(const float* __restrict__ part,
                                   float* __restrict__ attn_out) {
  const int g   = blockIdx.x;
  const int b   = blockIdx.y;
  const int tid = threadIdx.x;   // 128
  const size_t base = (size_t)(b * NKV + g) * NCHUNK * PART_STRIDE;

  float M = -INFINITY;
  for (int c = 0; c < NCHUNK; ++c) {
    const float lc = part[base + (size_t)c * PART_STRIDE + 1];
    if (lc > 0.0f) M = fmaxf(M, part[base + (size_t)c * PART_STRIDE]);
  }
  float L = 0.0f;
  float w[NCHUNK];
  for (int c = 0; c < NCHUNK; ++c) {
    const float mc = part[base + (size_t)c * PART_STRIDE];
    const float lc = part[base + (size_t)c * PART_STRIDE + 1];
    const float wc = (lc > 0.0f) ? __expf(mc - M) : 0.0f;
    w[c] = wc;
    L += lc * wc;
  }
  const float invL = 1.0f / L;

  for (int e = tid; e < HPG * HD; e += 128) {
    float o = 0.0f;
    for (int c = 0; c < NCHUNK; ++c)
      o += part[base + (size_t)c * PART_STRIDE + 2 + e] * w[c];
    attn_out[(size_t)b * DIM + g * (HPG * HD) + e] = o * invL;
  }
}

// ---------------------------------------------------------------------------
extern "C" void kernel_launch(void* const* d_in, const int* in_sizes, int n_in,
                              void* d_out, int out_size, void* d_ws, size_t ws_size,
                              hipStream_t stream) {
  (void)in_sizes; (void)n_in; (void)out_size; (void)ws_size;

  const float* x       = (const float*)d_in[0];
  const float* cache_k = (const float*)d_in[1];
  const float* cache_v = (const float*)d_in[2];
  const float* wqkv    = (const float*)d_in[3];
  const float* wo      = (const float*)d_in[4];
  const int*   ctx     = (const int*)d_in[5];
  float*       out     = (float*)d_out;

  // Workspace layout (floats), all regions 16B aligned. Total ~5.8 MB.
  float* ws    = (float*)d_ws;
  float* xqkv  = ws;                                            // 32*6144
  float* knew  = xqkv + (size_t)B * QKV_OUT;                    // 32*8*128
  float* vnew  = knew + (size_t)B * NKV * HD;                   // 32*8*128
  float* part  = vnew + (size_t)B * NKV * HD;                   // 32*8*8*520
  float* attno = part + (size_t)B * NKV * NCHUNK * PART_STRIDE; // 32*4096

  // 1) xqkv = x @ wqkv^T (zero first: split-K accumulates with f32 atomics)
  zero_f32<<<(B * QKV_OUT + 255) / 256, 256, 0, stream>>>(xqkv, B * QKV_OUT);
  gemm_xt_wmma<<<dim3(QKV_OUT / 64, SPLITK), 128, 0, stream>>>(
      x, wqkv, xqkv, QKV_OUT, DIM);

  // 2) RoPE q/k, stage new k,v row in scratch (inputs stay immutable)
  const int rope_total = B * NH * (HD / 2) + B * NKV * (HD / 2) + B * NKV * HD;
  rope_cache_kernel<<<(rope_total + 255) / 256, 256, 0, stream>>>(
      xqkv, knew, vnew, ctx);

  // 3) flash-decode partials over KV chunks (2048 blocks stream 1 GB of KV
  //    through the async global->LDS pipeline)
  attn_partial_kernel<<<dim3(NCHUNK, NKV, B), 256, 0, stream>>>(
      xqkv, cache_k, cache_v, knew, vnew, part, ctx);

  // 4) softmax cross-chunk reduction
  attn_reduce_kernel<<<dim3(NKV, B), 128, 0, stream>>>(part, attno);

  // 5) out = attn @ wo^T
  zero_f32<<<(B * DIM + 255) / 256, 256, 0, stream>>>(out, B * DIM);
  gemm_xt_wmma<<<dim3(DIM / 64, SPLITK), 128, 0, stream>>>(
      attno, wo, out, DIM, DIM);
}